// SDNN_LIFNeuron_13082470383701
// MI455X (gfx1250) — compile-verified
//
#include <hip/hip_runtime.h>
#include <cstdint>

// LIF scan over time axis for x[B=4, C=32, T=16, H=128, W=128] (f32).
//   y[t] = (1-alpha)*y[t-1] + x[t];  s[t] = (y[t] >= 1);  y[t] = 0 where spiked
// Memory-bound (268 MB traffic, ~11.5us @ 23.3 TB/s). Staging path uses the
// gfx1250 async global->LDS DMA so 16 b128 loads/lane are in flight with no
// VGPR destinations (ASYNCcnt-tracked), keeping register pressure minimal.

#define T_STEPS 16
#define HW      16384            // 128*128 elements per (b,c,t) slice
#define HW4     4096             // float4 slots per slice
#define PLANE_STRIDE (T_STEPS * HW)  // floats per (b,c) plane
#define BLOCK   128              // 4 waves; 32 KB LDS tile -> 10 blocks/WGP

__global__ __launch_bounds__(BLOCK) void lif_scan_kernel(
    const float* __restrict__ x, float* __restrict__ out) {
  __shared__ float4 tile[BLOCK * T_STEPS];  // 32 KB

  const int tid   = threadIdx.x;
  const int g     = blockIdx.x * BLOCK + tid;   // float4-column id
  const int plane = g >> 12;                    // g / HW4  (b*C + c)
  const int pos4  = g & (HW4 - 1);
  const long base = (long)plane * PLANE_STRIDE + (long)pos4 * 4;

  float4* myTile = &tile[tid * T_STEPS];

  // Stage the whole time column of this thread into LDS with async DMA.
  // Per wave: 16 x 512B coalesced transactions outstanding, zero dest VGPRs.
#pragma unroll
  for (int t = 0; t < T_STEPS; ++t) {
    const float* gptr = x + base + (long)t * HW;
    uint32_t lds_off = (uint32_t)(uintptr_t)(&myTile[t]);  // low 32b = LDS offset
    asm volatile("global_load_async_to_lds_b128 %0, %1, off"
                 :
                 : "v"(lds_off), "v"(gptr)
                 : "memory");
  }
  // Own-wave async writes to LDS are visible to this wave after the wait;
  // each lane reads only slots it wrote, so no workgroup barrier is needed.
  asm volatile("s_wait_asynccnt 0x0" ::: "memory");

  const float k = 0.05000000074505806f;  // float32(1.0 - 0.95), matches JAX
  float y0 = 0.f, y1 = 0.f, y2 = 0.f, y3 = 0.f;

#pragma unroll
  for (int t = 0; t < T_STEPS; ++t) {
    float4 v = myTile[t];                 // ds_load_b128
    y0 = k * y0 + v.x;
    y1 = k * y1 + v.y;
    y2 = k * y2 + v.z;
    y3 = k * y3 + v.w;
    float s0 = (y0 >= 1.0f) ? 1.0f : 0.0f;
    float s1 = (y1 >= 1.0f) ? 1.0f : 0.0f;
    float s2 = (y2 >= 1.0f) ? 1.0f : 0.0f;
    float s3 = (y3 >= 1.0f) ? 1.0f : 0.0f;
    y0 = (s0 != 0.0f) ? 0.0f : y0;        // reset where spiked
    y1 = (s1 != 0.0f) ? 0.0f : y1;
    y2 = (s2 != 0.0f) ? 0.0f : y2;
    y3 = (s3 != 0.0f) ? 0.0f : y3;
    float4 sv;
    sv.x = s0; sv.y = s1; sv.z = s2; sv.w = s3;
    *(float4*)(out + base + (long)t * HW) = sv;  // global_store_b128, coalesced
  }
}

extern "C" void kernel_launch(void* const* d_in, const int* in_sizes, int n_in,
                              void* d_out, int out_size, void* d_ws, size_t ws_size,
                              hipStream_t stream) {
  const float* x = (const float*)d_in[0];
  float* out = (float*)d_out;

  // planes = B*C derived from the input size so the mapping stays exact.
  const int planes  = in_sizes[0] / PLANE_STRIDE;   // 4*32 = 128
  const int columns = planes * HW4;                 // 524288 float4 columns
  const int blocks  = columns / BLOCK;              // 4096

  lif_scan_kernel<<<blocks, BLOCK, 0, stream>>>(x, out);
}